// CapsuleLayer_28114855920234
// MI455X (gfx1250) — compile-verified
//
#include <hip/hip_runtime.h>

// CapsuleLayer dynamic routing on MI455X (gfx1250, wave32).
// B=128, I=2048, P=8, J=32, D=16, 3 routing iterations.
// u_hat recomputed per routing pass with V_WMMA_F32_16X16X4_F32 (never hits HBM);
// x (8MB) + W (32MB) stay L2-resident.
// Tiles are computed TRANSPOSED (M=d, N=b) so the d-reductions of routing are
// register-direction FMAs + a single xor-16 ds_swizzle, not lane butterflies.

typedef float v2f __attribute__((ext_vector_type(2)));
typedef float v8f __attribute__((ext_vector_type(8)));

#define B_ 128
#define I_ 2048
#define P_ 8
#define J_ 32
#define D_ 16
#define EPS_ 1e-7f
#define ICHUNK 64   // i's per workgroup; grid.y = I_/ICHUNK = 32
#define LDSS 33     // padded LDS row stride (floats) -> conflict-free lane reads

// DPP lane-exchange within 16-lane groups (wave32). ctrl must be an immediate.
template <int CTRL>
__device__ __forceinline__ float dpp_xchg(float v) {
  return __int_as_float(__builtin_amdgcn_update_dpp(
      0, __float_as_int(v), CTRL, 0xf, 0xf, true));
}

// Butterfly sum/max over the 16 lanes of each half-wave.
__device__ __forceinline__ float groupSum16(float v) {
  v += dpp_xchg<0xB1>(v);   // quad_perm [1,0,3,2] : xor 1
  v += dpp_xchg<0x4E>(v);   // quad_perm [2,3,0,1] : xor 2
  v += dpp_xchg<0x141>(v);  // row_half_mirror     : xor 4
  v += dpp_xchg<0x140>(v);  // row_mirror          : xor 8
  return v;
}
__device__ __forceinline__ float groupMax16(float v) {
  v = fmaxf(v, dpp_xchg<0xB1>(v));
  v = fmaxf(v, dpp_xchg<0x4E>(v));
  v = fmaxf(v, dpp_xchg<0x141>(v));
  v = fmaxf(v, dpp_xchg<0x140>(v));
  return v;
}

// Exchange with lane^16 (crosses DPP16 row boundary): ds_swizzle xor-16.
__device__ __forceinline__ float xor16(float v) {
  return __int_as_float(
      __builtin_amdgcn_ds_swizzle(__float_as_int(v), 0x401F));  // xor=0x10,and=0x1f
}

__global__ void zero_kernel(float* __restrict__ p, int n) {
  int t = blockIdx.x * blockDim.x + threadIdx.x;
  if (t < n) p[t] = 0.0f;
}

// One fused routing pass over i.
// mode 0: c = 1/32 (softmax of zero logits); just accumulate s.
// mode 1: delta = v.u_hat; logits_out = delta; c = softmax_j(delta); s += c*u_hat.
// mode 2: delta = v.u_hat; c = softmax_j(logits_in + delta); s += c*u_hat.
// Grid: (B_/16, I_/ICHUNK). Block: 256 threads = 8 waves; wave w owns j in [4w,4w+4).
__global__ __launch_bounds__(256) void caps_route_kernel(
    const float* __restrict__ x,          // [B,I,P]
    const float* __restrict__ W,          // [J,I,D,P]
    const float* __restrict__ vprev,      // [B,J,D] (mode>=1)
    const float* __restrict__ logits_in,  // [B,I,J] (mode==2)
    float* __restrict__ logits_out,       // [B,I,J] (mode==1)
    float* __restrict__ s_out,            // [B,J,D] accumulated via atomics
    int mode) {
  __shared__ float lds_delta[16 * LDSS];
  __shared__ float lds_c[16 * LDSS];

  const int tid  = threadIdx.x;
  const int lane = tid & 31;
  const int wave = tid >> 5;
  const int hi   = lane >> 4;   // 0/1: K-half select; also d-register half (d=r+8*hi)
  const int lo   = lane & 15;   // A: M row = d; B/C/D: N col = b
  const int b0   = blockIdx.x * 16;
  const int i0   = blockIdx.y * ICHUNK;
  const int jb   = wave * 4;

  // v tiles matching transposed C-layout: vt[t][r] = v[b0+lo, jb+t, r + 8*hi]
  v8f vt[4];
  if (mode != 0) {
#pragma unroll
    for (int t = 0; t < 4; ++t) {
      const int j = jb + t;
      const float* vp = vprev + ((b0 + lo) * J_ + j) * D_ + 8 * hi;
      const float4 q0 = *(const float4*)(vp);
      const float4 q1 = *(const float4*)(vp + 4);
      vt[t][0] = q0.x; vt[t][1] = q0.y; vt[t][2] = q0.z; vt[t][3] = q0.w;
      vt[t][4] = q1.x; vt[t][5] = q1.y; vt[t][6] = q1.z; vt[t][7] = q1.w;
    }
  }

  // s accumulators, layout [d=r+8*hi][b=lo] per j-tile
  v8f sacc[4];
#pragma unroll
  for (int t = 0; t < 4; ++t)
#pragma unroll
    for (int r = 0; r < 8; ++r) sacc[t][r] = 0.0f;

  for (int i = i0; i < i0 + ICHUNK; ++i) {
    // ---- B fragment from x: x_i[p, b]; lane lo = b, K=p split across halves ----
    const float* xp = x + ((b0 + lo) * I_ + i) * P_;
    v2f xb0, xb1;
    {
      const float2 t0 = *(const float2*)(xp + 2 * hi);      // p = 2*hi, 2*hi+1
      const float2 t1 = *(const float2*)(xp + 4 + 2 * hi);  // p = 4+2*hi, 5+2*hi
      xb0.x = t0.x; xb0.y = t0.y;
      xb1.x = t1.x; xb1.y = t1.y;
    }

    // ---- u_hat^T tiles via WMMA f32 16x16x4: D[d,b] = W_ji[d,p] x x_i[p,b] ----
    v8f uh[4];
#pragma unroll
    for (int t = 0; t < 4; ++t) {
      const int j = jb + t;
      const float* wp = W + ((j * I_ + i) * D_ + lo) * P_;   // A: M=d=lo rows
      const float2 w0 = *(const float2*)(wp + 2 * hi);
      const float2 w1 = *(const float2*)(wp + 4 + 2 * hi);
      v2f wa0, wa1;
      wa0.x = w0.x; wa0.y = w0.y;
      wa1.x = w1.x; wa1.y = w1.y;
      v8f acc;
#pragma unroll
      for (int r = 0; r < 8; ++r) acc[r] = 0.0f;
      acc = __builtin_amdgcn_wmma_f32_16x16x4_f32(false, wa0, false, xb0,
                                                  (short)0, acc, false, false);
      acc = __builtin_amdgcn_wmma_f32_16x16x4_f32(false, wa1, false, xb1,
                                                  (short)0, acc, false, false);
      uh[t] = acc;
    }

    if (mode == 0) {
      // uniform coupling c = 1/J
#pragma unroll
      for (int t = 0; t < 4; ++t)
#pragma unroll
        for (int r = 0; r < 8; ++r) sacc[t][r] += (1.0f / 32.0f) * uh[t][r];
    } else {
      // ---- delta[b,j] = sum_d v*u_hat: 8 FMAs over regs + one xor16 hop ----
#pragma unroll
      for (int t = 0; t < 4; ++t) {
        const int j = jb + t;
        float tsum = uh[t][0] * vt[t][0];
#pragma unroll
        for (int r = 1; r < 8; ++r) tsum = fmaf(uh[t][r], vt[t][r], tsum);
        tsum += xor16(tsum);                 // combine d=0..7 with d=8..15 halves
        if (hi == 0) lds_delta[lo * LDSS + j] = tsum;  // lane lo = b
      }
      __syncthreads();

      // ---- softmax over j (32 values per (b,i)); 16 threads/row x 2 j each ----
      {
        const int bb = tid >> 4;
        const int jj = tid & 15;
        float d0 = lds_delta[bb * LDSS + jj];
        float d1 = lds_delta[bb * LDSS + jj + 16];
        const int lbase = ((b0 + bb) * I_ + i) * J_;
        if (mode == 2) {
          d0 += logits_in[lbase + jj];
          d1 += logits_in[lbase + jj + 16];
        }
        if (mode == 1) {
          logits_out[lbase + jj]      = d0;
          logits_out[lbase + jj + 16] = d1;
        }
        float m = groupMax16(fmaxf(d0, d1));
        float e0 = __expf(d0 - m);
        float e1 = __expf(d1 - m);
        float denom = groupSum16(e0 + e1);
        float inv = 1.0f / denom;
        lds_c[bb * LDSS + jj]      = e0 * inv;
        lds_c[bb * LDSS + jj + 16] = e1 * inv;
      }
      __syncthreads();

      // ---- s += c * u_hat : one broadcast c per tile, then register FMAs ----
#pragma unroll
      for (int t = 0; t < 4; ++t) {
        const int j = jb + t;
        const float cc = lds_c[lo * LDSS + j];   // lane lo = b
#pragma unroll
        for (int r = 0; r < 8; ++r) sacc[t][r] = fmaf(cc, uh[t][r], sacc[t][r]);
      }
    }
  }

  // ---- reduce i-chunks: native f32 atomics into s_out[B,J,D] ----
#pragma unroll
  for (int t = 0; t < 4; ++t) {
    const int j = jb + t;
#pragma unroll
    for (int r = 0; r < 8; ++r) {
      unsafeAtomicAdd(&s_out[((b0 + lo) * J_ + j) * D_ + r + 8 * hi], sacc[t][r]);
    }
  }
}

// squash over D=16 (one 16-lane group per (b,j) row). Grid is exact: B*J*D threads.
__global__ __launch_bounds__(256) void squash_kernel(const float* __restrict__ s,
                                                     float* __restrict__ out) {
  const int t = blockIdx.x * blockDim.x + threadIdx.x;
  const float val = s[t];
  const float s2 = groupSum16(val * val);
  const float scale = (s2 / (1.0f + s2)) * rsqrtf(s2 + EPS_);
  out[t] = val * scale;
}

extern "C" void kernel_launch(void* const* d_in, const int* in_sizes, int n_in,
                              void* d_out, int out_size, void* d_ws, size_t ws_size,
                              hipStream_t stream) {
  const float* x = (const float*)d_in[0];  // [128,2048,8]
  const float* W = (const float*)d_in[1];  // [32,2048,16,8]
  float* out = (float*)d_out;              // [128,32,16]
  float* ws = (float*)d_ws;

  const int S = B_ * J_ * D_;  // 65536
  float* s0 = ws;
  float* s1 = ws + S;
  float* s2 = ws + 2 * S;
  float* v0 = ws + 3 * S;
  float* v1 = ws + 4 * S;
  float* logits = ws + 5 * S;  // [B,I,J] = 8,388,608 floats (~32 MB)
  // total ws use: ~33.3 MB

  dim3 grid(B_ / 16, I_ / ICHUNK);

  zero_kernel<<<(3 * S + 255) / 256, 256, 0, stream>>>(s0, 3 * S);

  // routing iter 0: uniform c
  caps_route_kernel<<<grid, 256, 0, stream>>>(x, W, nullptr, nullptr, nullptr, s0, 0);
  squash_kernel<<<S / 256, 256, 0, stream>>>(s0, v0);

  // routing iter 1: logits1 = v0.u_hat; c = softmax(logits1)
  caps_route_kernel<<<grid, 256, 0, stream>>>(x, W, v0, nullptr, logits, s1, 1);
  squash_kernel<<<S / 256, 256, 0, stream>>>(s1, v1);

  // routing iter 2 (final): c = softmax(logits1 + v1.u_hat)
  caps_route_kernel<<<grid, 256, 0, stream>>>(x, W, v1, logits, nullptr, s2, 2);
  squash_kernel<<<S / 256, 256, 0, stream>>>(s2, out);
}